// SimpleGCN_9852654977192
// MI455X (gfx1250) — compile-verified
//
#include <hip/hip_runtime.h>

#define NN 100000
#define NE 1600000

typedef __attribute__((ext_vector_type(2))) float v2f;
typedef __attribute__((ext_vector_type(8))) float v8f;

// ---------- utility fills (graph-capture-safe, no hipMemset*) ----------
__global__ __launch_bounds__(256) void k_fill(float* __restrict__ p, float v, long n) {
  long i = (long)blockIdx.x * 256 + threadIdx.x;
  if (i < n) p[i] = v;
}

// ---------- degree: deg starts at 1.0 (self loop), +1 per incoming edge ----------
__global__ __launch_bounds__(256) void k_deg(const long long* __restrict__ ei,
                                             float* __restrict__ deg) {
  int e = blockIdx.x * 256 + threadIdx.x;
  if (e < NE) atomicAdd(&deg[(int)ei[NE + e]], 1.0f);
}

__global__ __launch_bounds__(256) void k_rsqrt(float* __restrict__ d, int n) {
  int i = blockIdx.x * 256 + threadIdx.x;
  if (i < n) d[i] = rsqrtf(d[i]);
}

// ---------- fp32 WMMA GEMM: C[M,N] = A[M,K] * B[K,N], one wave per 16-row strip ----------
// A-fragment layout (32-bit A 16x4): lanes 0-15 -> M=lane, K=k0..k0+1 in v[0..1];
//                                    lanes 16-31 -> M=lane-16, K=k0+2..k0+3.
// B/C/D: row striped across lanes (lanes 0-15 N=0..15, lanes 16-31 repeat cols,
//        C/D VGPR r -> M=r (lanes 0-15) / M=r+8 (lanes 16-31)).
template <int NT>  // NT = N/16 column tiles (8 for N=128, 4 for N=64)
__global__ __launch_bounds__(128) void k_gemm(const float* __restrict__ A,
                                              const float* __restrict__ B,
                                              float* __restrict__ C, int M, int K) {
  const int tile = blockIdx.x * 4 + (threadIdx.x >> 5);
  if (tile * 16 >= M) return;  // whole-wave exit: EXEC stays all-ones for WMMA
  const int lane = threadIdx.x & 31;
  const int half = lane >> 4;
  const int l15  = lane & 15;
  const int N    = NT * 16;
  const int m0   = tile * 16;

  v8f zero = {};
  v8f acc[NT];
#pragma unroll
  for (int t = 0; t < NT; ++t) acc[t] = zero;

  const float* arow = A + (size_t)(m0 + l15) * K;
  for (int k0 = 0; k0 < K; k0 += 4) {
    const int ka = k0 + half * 2;
    v2f a = *(const v2f*)(arow + ka);  // 8B-aligned: ka even
#pragma unroll
    for (int t = 0; t < NT; ++t) {
      v2f b;
      b.x = B[ka * N + t * 16 + l15];
      b.y = B[(ka + 1) * N + t * 16 + l15];
      acc[t] = __builtin_amdgcn_wmma_f32_16x16x4_f32(
          /*neg_a=*/false, a, /*neg_b=*/false, b,
          /*c_mod=*/(short)0, acc[t], /*reuse_a=*/false, /*reuse_b=*/false);
    }
  }
#pragma unroll
  for (int t = 0; t < NT; ++t) {
#pragma unroll
    for (int r = 0; r < 8; ++r) {
      C[(size_t)(m0 + r + half * 8) * N + t * 16 + l15] = acc[t][r];
    }
  }
}

// ---------- edge scatter: agg[dst] += h[src] * dinv[src]*dinv[dst] ----------
template <int CH>  // 128 -> float4 per lane, 64 -> float2 per lane
__global__ __launch_bounds__(256) void k_scatter(const long long* __restrict__ ei,
                                                 const float* __restrict__ h,
                                                 const float* __restrict__ dinv,
                                                 float* __restrict__ agg) {
  const int e = blockIdx.x * 8 + (threadIdx.x >> 5);  // one wave per edge
  if (e >= NE) return;
  const int lane = threadIdx.x & 31;
  const int s = (int)ei[e];
  const int d = (int)ei[NE + e];
  const float w = dinv[s] * dinv[d];
  constexpr int PER = CH / 32;
  const float* hp = h + (size_t)s * CH + lane * PER;
  float* ap = agg + (size_t)d * CH + lane * PER;
  if constexpr (PER == 4) {
    float4 v = *(const float4*)hp;  // GLOBAL_LOAD_B128, 512B coalesced per wave
    atomicAdd(ap + 0, v.x * w);
    atomicAdd(ap + 1, v.y * w);
    atomicAdd(ap + 2, v.z * w);
    atomicAdd(ap + 3, v.w * w);
  } else {
    float2 v = *(const float2*)hp;
    atomicAdd(ap + 0, v.x * w);
    atomicAdd(ap + 1, v.y * w);
  }
}

// ---------- self-loop term + bias (+ optional relu), in place on agg ----------
template <int CH, bool RELU>
__global__ __launch_bounds__(256) void k_finalize(const float* __restrict__ h,
                                                  const float* __restrict__ dinv,
                                                  const float* __restrict__ bias,
                                                  float* __restrict__ agg) {
  const long i = (long)blockIdx.x * 256 + threadIdx.x;
  if (i >= (long)NN * CH) return;
  const int node = (int)(i / CH);
  const int c    = (int)(i & (CH - 1));
  const float di = dinv[node];
  float v = agg[i] + h[i] * di * di + bias[c];
  agg[i] = RELU ? fmaxf(v, 0.0f) : v;
}

extern "C" void kernel_launch(void* const* d_in, const int* in_sizes, int n_in,
                              void* d_out, int out_size, void* d_ws, size_t ws_size,
                              hipStream_t stream) {
  const float*     x  = (const float*)d_in[0];      // [NN,128]
  const float*     W1 = (const float*)d_in[1];      // [128,128]
  const float*     b1 = (const float*)d_in[2];      // [128]
  const float*     W2 = (const float*)d_in[3];      // [128,64]
  const float*     b2 = (const float*)d_in[4];      // [64]
  const long long* ei = (const long long*)d_in[5];  // [2,NE] int64
  float* out = (float*)d_out;                       // [NN,64]

  // workspace carve-up (floats): dinv | h1 (reused as h2) | agg1
  float* dinv = (float*)d_ws;                      // NN
  float* h1   = dinv + ((NN + 255) & ~255);        // NN*128
  float* agg1 = h1 + (size_t)NN * 128;             // NN*128
  float* h2   = h1;                                // NN*64 overlays h1 (dead by then)

  const long n_h1  = (long)NN * 128;
  const long n_out = (long)NN * 64;

  // degree / norm
  k_fill<<<(NN + 255) / 256, 256, 0, stream>>>(dinv, 1.0f, NN);
  k_fill<<<(int)((n_h1 + 255) / 256), 256, 0, stream>>>(agg1, 0.0f, n_h1);
  k_fill<<<(int)((n_out + 255) / 256), 256, 0, stream>>>(out, 0.0f, n_out);
  k_deg<<<(NE + 255) / 256, 256, 0, stream>>>(ei, dinv);
  k_rsqrt<<<(NN + 255) / 256, 256, 0, stream>>>(dinv, NN);

  const int gemm_grid = (NN / 16 + 3) / 4;  // 6250 tiles, 4 waves/block

  // layer 1: h1 = x @ W1 ; agg1 = scatter + self + b1 ; relu
  k_gemm<8><<<gemm_grid, 128, 0, stream>>>(x, W1, h1, NN, 128);
  k_scatter<128><<<(NE + 7) / 8, 256, 0, stream>>>(ei, h1, dinv, agg1);
  k_finalize<128, true><<<(int)((n_h1 + 255) / 256), 256, 0, stream>>>(h1, dinv, b1, agg1);

  // layer 2: h2 = agg1 @ W2 ; out = scatter + self + b2
  k_gemm<4><<<gemm_grid, 128, 0, stream>>>(agg1, W2, h2, NN, 128);
  k_scatter<64><<<(NE + 7) / 8, 256, 0, stream>>>(ei, h2, dinv, out);
  k_finalize<64, false><<<(int)((n_out + 255) / 256), 256, 0, stream>>>(h2, dinv, b2, out);
}